// GeodesicEvolution_60619168416505
// MI455X (gfx1250) — compile-verified
//
#include <hip/hip_runtime.h>
#include <hip/hip_bf16.h>

// CDNA5 (gfx1250) wave32 WMMA types
typedef __attribute__((ext_vector_type(2))) float v2f;
typedef __attribute__((ext_vector_type(8))) float v8f;

// One block = one token (b,t) x one quarter of the k-range (16 k values).
// blockDim = 256 (8 wave32); each wave emits 2 k-slabs of Gamma via
// v_wmma_f32_16x16x4_f32 rank-2 outer-product tiles + NT streaming stores.
extern "C" __global__ void __launch_bounds__(256)
geodesic_gamma_wmma(const float* __restrict__ zg,
                    const float* __restrict__ vg,
                    const float* __restrict__ Lam,
                    const float* __restrict__ alpha_p,
                    float* __restrict__ out,
                    int ntok)
{
    constexpr int D = 64;
    const int bt   = blockIdx.x;      // token index in [0, B*T)
    const int kblk = blockIdx.y;      // which 16 k-rows this block owns
    const int tid  = threadIdx.x;
    const float alpha = alpha_p[0];

    __shared__ float z_sh[D];       // z_i
    __shared__ float zh_sh[D];      // zhat_i = z_i / d_i
    __shared__ float ainvd_sh[D];   // alpha_eff / d_k          (diag part of alpha*G row k)
    __shared__ float c_sh[D];       // alpha_eff * s * zhat_k   (rank-1 part of alpha*G row k)
    __shared__ float aw_sh[D];      // alpha_eff * w_k          (delta_ij term)
    __shared__ float part[2][8];    // cross-wave reduction scratch

    // ---- setup: per-token Sherman-Morrison quantities (waves 0-1 only) ----
    float zk = 0.f, vk = 0.f, dk = 0.f, zh = 0.f;
    if (tid < D) {
        zk = zg[(size_t)bt * D + tid];
        vk = vg[(size_t)bt * D + tid];
        dk = Lam[tid];                       // stash Lambda_k in dk for now
        float p0 = zk * zk, p1 = zk * vk, p2 = vk * vk;
        #pragma unroll
        for (int m = 16; m >= 1; m >>= 1) {
            p0 += __shfl_xor(p0, m, 32);
            p1 += __shfl_xor(p1, m, 32);
            p2 += __shfl_xor(p2, m, 32);
        }
        if ((tid & 31) == 0) {
            const int wv = tid >> 5;
            part[wv][0] = p0; part[wv][1] = p1; part[wv][2] = p2;
        }
    }
    __syncthreads();
    float zdotv = 0.f, vdotv = 0.f;
    if (tid < D) {
        const float z2 = part[0][0] + part[1][0];
        zdotv = part[0][1] + part[1][1];
        vdotv = part[0][2] + part[1][2];
        dk = dk + alpha * z2 + 1e-6f;        // d_k = Lambda_k + alpha|z|^2 + 1e-6
        zh = zk / dk;
        float p0 = zk * zh, p1 = vk * zh;    // -> q = z.zhat, r = v.zhat
        #pragma unroll
        for (int m = 16; m >= 1; m >>= 1) {
            p0 += __shfl_xor(p0, m, 32);
            p1 += __shfl_xor(p1, m, 32);
        }
        if ((tid & 31) == 0) {
            const int wv = tid >> 5;
            part[wv][4] = p0; part[wv][5] = p1;
        }
    }
    __syncthreads();
    if (tid < D) {
        const float q      = part[0][4] + part[1][4];
        const float r      = part[0][5] + part[1][5];
        const float invden = 1.0f / (1.0f + 2.0f * alpha * q);
        const float s      = 2.0f * alpha * invden;
        const float aeff   = (alpha < 1e-8f) ? 0.0f : alpha;  // zero-branch of reference

        z_sh[tid]     = zk;
        zh_sh[tid]    = zh;
        ainvd_sh[tid] = aeff / dk;
        c_sh[tid]     = aeff * s * zh;
        aw_sh[tid]    = aeff * zh * invden;                   // aeff * w_k

        if (kblk == 0) {
            // a_geo_k = -alpha * (2 (z.v) u_k + |v|^2 w_k), u = G v
            const float u = vk / dk - s * zh * r;
            const float w = zh * invden;
            out[(size_t)ntok * D * D * D + (size_t)bt * D + tid] =
                -aeff * (2.0f * zdotv * u + vdotv * w);
        }
    }
    __syncthreads();

    // ---- main: Gamma[k,i,j] = (aG_k)_i z_j + z_i (aG_k)_j + delta_ij aw_k ----
    const int wave = tid >> 5;
    const int lane = tid & 31;
    const int lo   = lane & 15;
    const int hi   = lane >> 4;
    const float hm = hi ? 0.0f : 1.0f;   // lanes 16-31 carry K=2,3 == 0
    float* const gbase = out + (size_t)bt * (D * D * D);

    #pragma unroll
    for (int kk = 0; kk < 2; ++kk) {
        const int k = kblk * 16 + wave * 2 + kk;
        const float ainvd = ainvd_sh[k];
        const float ck    = c_sh[k];
        const float awk   = aw_sh[k];
        float* const gk = gbase + (size_t)k * (D * D);

        #pragma unroll
        for (int ti = 0; ti < 4; ++ti) {
            const int i = ti * 16 + lo;
            // A (16x4): col0 = alpha*G[k,i], col1 = z_i, col2..3 = 0
            const float aGi = ((i == k) ? ainvd : 0.0f) - ck * zh_sh[i];
            v2f A = { hm * aGi, hm * z_sh[i] };

            #pragma unroll
            for (int tj = 0; tj < 4; ++tj) {
                const int j = tj * 16 + lo;
                // B (4x16): row0 = z_j, row1 = alpha*G[k,j], row2..3 = 0
                const float aGj = ((j == k) ? ainvd : 0.0f) - ck * zh_sh[j];
                v2f Bm = { hm * z_sh[j], hm * aGj };

                v8f C = {0.f, 0.f, 0.f, 0.f, 0.f, 0.f, 0.f, 0.f};
                if (ti == tj) {  // diagonal tile: C[m,n] = (m==n) * (alpha*w_k)
                    #pragma unroll
                    for (int r = 0; r < 8; ++r)
                        C[r] = (lo == r + 8 * hi) ? awk : 0.0f;
                }

                // D = A x B + C  ==  aG_k (x) z + z (x) aG_k + diag(aw_k)
                v8f Dv = __builtin_amdgcn_wmma_f32_16x16x4_f32(
                    false, A, false, Bm, (short)0, C, false, false);

                // lane holds rows m = r + 8*hi, col n = lo of this 16x16 tile
                float* gp = gk + (size_t)(ti * 16 + 8 * hi) * D + tj * 16 + lo;
                #pragma unroll
                for (int r = 0; r < 8; ++r)
                    __builtin_nontemporal_store(Dv[r], gp + (size_t)r * D);
            }
        }
    }
}

extern "C" void kernel_launch(void* const* d_in, const int* in_sizes, int n_in,
                              void* d_out, int out_size, void* d_ws, size_t ws_size,
                              hipStream_t stream) {
    const float* z     = (const float*)d_in[0];   // (B,T,64)
    const float* v     = (const float*)d_in[1];   // (B,T,64)
    const float* Lam   = (const float*)d_in[2];   // (64,)
    const float* alpha = (const float*)d_in[3];   // scalar
    float* out = (float*)d_out;                   // Gamma (B,T,64,64,64) ++ a_geo (B,T,64)

    const int ntok = in_sizes[0] / 64;            // B*T = 256
    dim3 grid(ntok, 4, 1);                        // 4 k-blocks of 16 rows each
    geodesic_gamma_wmma<<<grid, 256, 0, stream>>>(z, v, Lam, alpha, out, ntok);
}